// ProjectLayer_53145925321507
// MI455X (gfx1250) — compile-verified
//
#include <hip/hip_runtime.h>

#define IMG_WF 512.0f
#define HM_W 128
#define HM_HW (HM_W * HM_W)   // 16384
#define NV 5
#define NB 4
#define NJ 15
#define P_TOT (64 * 64 * 64)  // 262144 voxels per batch

typedef float v2f __attribute__((ext_vector_type(2)));
typedef float v8f __attribute__((ext_vector_type(8)));

__global__ __launch_bounds__(256) void project_voxel_kernel(
    const float* __restrict__ hm,       // (NV, NB, NJ, 128, 128)
    const float* __restrict__ projM,    // (NV, NB, 3, 4)
    const float* __restrict__ center,   // (NB, 3)
    const int*   __restrict__ gsize,    // scalar
    float* __restrict__ cubes,          // (NB, NJ, P_TOT)
    float* __restrict__ grids)          // (NB, P_TOT, 3)
{
  // Per-wave private LDS transpose buffers: 8 waves x 2 tiles x 16x16 (+pad)
  __shared__ float lds[8][2][16 * 17];

  const int tid  = threadIdx.x;
  const int lane = tid & 31;
  const int wave = tid >> 5;
  const int b    = blockIdx.y;
  const int base = blockIdx.x * 256 + wave * 32;   // wave's first voxel (within batch)

  const float gs   = (float)gsize[0];
  const float half = gs * 0.5f;
  const float step = gs / 63.0f;
  const float cx = center[b * 3 + 0];
  const float cy = center[b * 3 + 1];
  const float cz = center[b * 3 + 2];

  auto coords = [&](int p, float& gx, float& gy, float& gz) {
    int iz = p & 63, iy = (p >> 6) & 63, ix = p >> 12;
    gx = -half + step * (float)ix + cx;
    gy = -half + step * (float)iy + cy;
    gz = -half + step * (float)iz + cz;
  };

  const int  col    = lane & 15;
  const bool hiHalf = lane >= 16;

  // Coordinates for WMMA tile 0 (points base..base+15) and tile 1 (base+16..base+31)
  float gx0, gy0, gz0, gx1, gy1, gz1;
  coords(base + col,      gx0, gy0, gz0);
  coords(base + 16 + col, gx1, gy1, gz1);

  // This lane's own voxel (for grids output + sampling phase)
  const int p_me = base + lane;
  const float mx = hiHalf ? gx1 : gx0;
  const float my = hiHalf ? gy1 : gy0;
  const float mz = hiHalf ? gz1 : gz0;
  {
    size_t gb = ((size_t)b * P_TOT + (size_t)p_me) * 3;
    grids[gb + 0] = mx;
    grids[gb + 1] = my;
    grids[gb + 2] = mz;
  }

  float PX[NV], PY[NV], PZ[NV];

#if defined(__gfx1250__) && __has_builtin(__builtin_amdgcn_wmma_f32_16x16x4_f32)
  // ---- Projection via matrix core: D(16x16) = A(16 pts x 4 homog) * B(4 x 16 cams) ----
  // homog order (K) = (gx, gz, gy, 1)  [pts permuted [0,2,1] in reference]
  // A layout (f32 16x4): lanes 0-15 -> {K0,K1}, lanes 16-31 -> {K2,K3}, M = lane&15
  v2f a0, a1, bb;
  if (!hiHalf) { a0.x = gx0; a0.y = gz0; a1.x = gx1; a1.y = gz1; }
  else         { a0.x = gy0; a0.y = 1.f; a1.x = gy1; a1.y = 1.f; }
  // B column l = (view n = l/3, proj row i = l%3) for l < 15, zero otherwise
  bb.x = 0.f; bb.y = 0.f;
  if (col < NV * 3) {
    int n = col / 3, i = col - 3 * n;
    const float* M = projM + (((size_t)n * NB + b) * 3 + i) * 4;
    bb.x = hiHalf ? M[2] : M[0];
    bb.y = hiHalf ? M[3] : M[1];
  }
  v8f c0 = {};
  v8f d0 = __builtin_amdgcn_wmma_f32_16x16x4_f32(false, a0, false, bb, (short)0, c0, false, false);
  v8f d1 = __builtin_amdgcn_wmma_f32_16x16x4_f32(false, a1, false, bb, (short)0, c0, false, false);

  // Transpose D through wave-private LDS: row = point-in-tile, col = 3*view+comp.
  // Row stride 17 floats avoids the 2-way bank conflict between the lane halves.
  {
    const int rowoff = hiHalf ? 8 : 0;
#pragma unroll
    for (int r = 0; r < 8; ++r) {
      lds[wave][0][(r + rowoff) * 17 + col] = d0[r];
      lds[wave][1][(r + rowoff) * 17 + col] = d1[r];
    }
  }
  asm volatile("s_wait_dscnt 0" ::: "memory");   // wave-internal RAW through LDS
  {
    const float* row = &lds[wave][hiHalf ? 1 : 0][col * 17];
#pragma unroll
    for (int n = 0; n < NV; ++n) {
      PX[n] = row[3 * n + 0];
      PY[n] = row[3 * n + 1];
      PZ[n] = row[3 * n + 2];
    }
  }
#else
  // Scalar fallback projection
#pragma unroll
  for (int n = 0; n < NV; ++n) {
    const float* M = projM + ((size_t)n * NB + b) * 12;
    PX[n] = M[0] * mx + M[1] * mz + M[2]  * my + M[3];
    PY[n] = M[4] * mx + M[5] * mz + M[6]  * my + M[7];
    PZ[n] = M[8] * mx + M[9] * mz + M[10] * my + M[11];
  }
#endif

  // ---- Gather/bilinear phase: L2-resident heatmaps, 300 b32 gathers per voxel ----
  float acc[NJ];
#pragma unroll
  for (int j = 0; j < NJ; ++j) acc[j] = 0.f;
  float den = 0.f;

#pragma unroll
  for (int n = 0; n < NV; ++n) {
    // Fast perspective divide: one v_rcp_f32 + two multiplies instead of two
    // IEEE division macro-sequences. Z==0 -> inf/NaN coords -> inb fails, same
    // as the reference's semantics. Keeps address generation off the critical
    // divide-refinement chain so the gather clause issues sooner.
    const float rz = __builtin_amdgcn_rcpf(PZ[n]);
    const float xr = PX[n] * rz;
    const float yr = PY[n] * rz;
    const bool inb = (xr >= 0.f) && (yr >= 0.f) && (xr < IMG_WF) && (yr < IMG_WF);
    if (inb) {
      // replicate reference chain: clip(xy,-1,512) * (128/512) -> sg -> unnormalize
      float xc  = fminf(fmaxf(xr, -1.f), 512.f) * 0.25f;
      float yc  = fminf(fmaxf(yr, -1.f), 512.f) * 0.25f;
      float sgx = fminf(fmaxf(xc * (2.0f / 127.0f) - 1.0f, -1.1f), 1.1f);
      float sgy = fminf(fmaxf(yc * (2.0f / 127.0f) - 1.0f, -1.1f), 1.1f);
      float ixf = (sgx + 1.f) * 63.5f;
      float iyf = (sgy + 1.f) * 63.5f;
      float x0f = floorf(ixf), y0f = floorf(iyf);
      float wx1 = ixf - x0f, wx0 = 1.f - wx1;
      float wy1 = iyf - y0f, wy0 = 1.f - wy1;
      float vx0 = (x0f >= 0.f       && x0f <= 127.f)       ? 1.f : 0.f;
      float vx1 = (x0f + 1.f >= 0.f && x0f + 1.f <= 127.f) ? 1.f : 0.f;
      float vy0 = (y0f >= 0.f       && y0f <= 127.f)       ? 1.f : 0.f;
      float vy1 = (y0f + 1.f >= 0.f && y0f + 1.f <= 127.f) ? 1.f : 0.f;
      int x0 = min(max((int)x0f, 0), 127);
      int x1 = min(max((int)x0f + 1, 0), 127);
      int y0 = min(max((int)y0f, 0), 127);
      int y1 = min(max((int)y0f + 1, 0), 127);
      float w00 = wx0 * wy0 * vx0 * vy0;
      float w01 = wx1 * wy0 * vx1 * vy0;
      float w10 = wx0 * wy1 * vx0 * vy1;
      float w11 = wx1 * wy1 * vx1 * vy1;
      int o00 = y0 * HM_W + x0, o01 = y0 * HM_W + x1;
      int o10 = y1 * HM_W + x0, o11 = y1 * HM_W + x1;
      const float* hb = hm + ((size_t)n * NB + b) * NJ * HM_HW;
#pragma unroll
      for (int j = 0; j < NJ; ++j) {
        const float* h = hb + (size_t)j * HM_HW;
        acc[j] += h[o00] * w00 + h[o01] * w01 + h[o10] * w10 + h[o11] * w11;
      }
      den += 1.f;
    }
  }

  const float inv = 1.f / (den + 1e-6f);
#pragma unroll
  for (int j = 0; j < NJ; ++j) {
    float v = acc[j] * inv;
    v = (v != v) ? 0.f : v;                 // NaN -> 0 per reference
    v = fminf(fmaxf(v, 0.f), 1.f);
    cubes[((size_t)b * NJ + j) * P_TOT + (size_t)p_me] = v;
  }
}

extern "C" void kernel_launch(void* const* d_in, const int* in_sizes, int n_in,
                              void* d_out, int out_size, void* d_ws, size_t ws_size,
                              hipStream_t stream) {
  const float* hm     = (const float*)d_in[0];  // heatmaps
  const float* projM  = (const float*)d_in[1];
  const float* center = (const float*)d_in[2];
  const int*   gs     = (const int*)d_in[3];    // grid_size scalar

  float* cubes = (float*)d_out;
  float* grids = cubes + (size_t)NB * NJ * P_TOT;   // outputs concatenated flat

  dim3 grid(P_TOT / 256, NB);   // (1024, 4) blocks x 256 threads, one thread per voxel
  project_voxel_kernel<<<grid, 256, 0, stream>>>(hm, projM, center, gs, cubes, grids);
}